// GINENet_41532333752774
// MI455X (gfx1250) — compile-verified
//
#include <hip/hip_runtime.h>

typedef __attribute__((ext_vector_type(2))) float v2f;
typedef __attribute__((ext_vector_type(8))) float v8f;

#define NNODES 50000
#define NEDGES 400000
#define FNODE  64
#define FEDGE  16
#define HDIM   256
#define NGRAPH 256
#define BN_EPS 1e-5f

// ---------------------------------------------------------------------------
// CDNA5 async global->LDS copy (16B per lane), ASYNCcnt-tracked.
// LDS operand = low 32 bits of the generic __shared__ pointer
// (generic LDS addr = {aperture, lds_offset}; HW adds LDS_BASE).
// ---------------------------------------------------------------------------
__device__ __forceinline__ void async_ld_b128(float* lp, const float* gp)
{
    unsigned l = (unsigned)(unsigned long long)lp;
    asm volatile("global_load_async_to_lds_b128 %0, %1, off"
                 :: "v"(l), "v"(gp) : "memory");
}
__device__ __forceinline__ void wait_async_le2()
{
    asm volatile("s_wait_asynccnt 2" ::: "memory");
}
__device__ __forceinline__ void wait_async_0()
{
    asm volatile("s_wait_asynccnt 0" ::: "memory");
}

// ---------------------------------------------------------------------------
// fp32 WMMA GEMM:  C[M,N] = f( A[M,K] @ W[K,N] + bias )
// Block = 256 threads (8 wave32), tile 64(M) x 64(N), K step 16.
// Each wave computes a 16x32 slice via two v_wmma_f32_16x16x4_f32 accums.
// BN=false: double-buffered GLOBAL_LOAD_ASYNC_TO_LDS_B128 staging.
// BN=true : synchronous staging with per-K BN(scale,shift)+ReLU fused on A.
// OOB A rows are clamped (they only feed OOB C rows, masked at store).
// ---------------------------------------------------------------------------
template <bool BN>
__global__ __launch_bounds__(256)
void gemm_wmma_kernel(const float* __restrict__ A, const float* __restrict__ W,
                      const float* __restrict__ bias,
                      const float* __restrict__ bnScale, const float* __restrict__ bnShift,
                      float* __restrict__ C, int M, int K, int N, int doRelu)
{
    __shared__ float As[2][64 * 17];   // padded stride 17
    __shared__ float Ws[2][16 * 65];   // padded stride 65

    const int tid  = threadIdx.x;
    const int lane = tid & 31;
    const int wid  = tid >> 5;
    const int l    = lane & 15;
    const int hi   = lane >> 4;

    const int mBlock = blockIdx.y * 64;
    const int nBlock = blockIdx.x * 64;
    const int mW = (wid & 3) * 16;
    const int nW = (wid >> 2) * 32;

    v8f c0 = {};
    v8f c1 = {};

    // cooperative tile-load coordinates (each thread owns one float4 group)
    const int ar = tid >> 2;        // 0..63 (A tile row)
    const int ac = (tid & 3) * 4;   // 0,4,8,12
    const int wr = tid >> 4;        // 0..15 (W tile row)
    const int wc = (tid & 15) * 4;  // 0..60

    int arow = mBlock + ar;
    if (arow >= M) arow = M - 1;    // clamp: OOB A rows are dead (see header)
    const float* Abase = A + (size_t)arow * K + ac;
    const float* Wbase = W + (size_t)wr * N + nBlock + wc;

    const int nt = K >> 4;

    if (!BN) {
        // prefetch tile 0
        async_ld_b128(&As[0][ar * 17 + ac], Abase);
        async_ld_b128(&Ws[0][wr * 65 + wc], Wbase);
    }

    for (int i = 0; i < nt; ++i) {
        const int cur = i & 1;
        const int kt  = i << 4;

        if (!BN) {
            if (i + 1 < nt) {
                // issue next tile, then wait for current tile (in-order)
                async_ld_b128(&As[cur ^ 1][ar * 17 + ac], Abase + kt + 16);
                async_ld_b128(&Ws[cur ^ 1][wr * 65 + wc],
                              Wbase + (size_t)(kt + 16) * N);
                wait_async_le2();
            } else {
                wait_async_0();
            }
        } else {
            // synchronous staging with fused BN + ReLU on the A input
            float4 av = *(const float4*)(Abase + kt);
            const int k0 = kt + ac;
            av.x = fmaxf(av.x * bnScale[k0 + 0] + bnShift[k0 + 0], 0.f);
            av.y = fmaxf(av.y * bnScale[k0 + 1] + bnShift[k0 + 1], 0.f);
            av.z = fmaxf(av.z * bnScale[k0 + 2] + bnShift[k0 + 2], 0.f);
            av.w = fmaxf(av.w * bnScale[k0 + 3] + bnShift[k0 + 3], 0.f);
            As[cur][ar * 17 + ac + 0] = av.x;
            As[cur][ar * 17 + ac + 1] = av.y;
            As[cur][ar * 17 + ac + 2] = av.z;
            As[cur][ar * 17 + ac + 3] = av.w;

            float4 wv = *(const float4*)(Wbase + (size_t)kt * N);
            Ws[cur][wr * 65 + wc + 0] = wv.x;
            Ws[cur][wr * 65 + wc + 1] = wv.y;
            Ws[cur][wr * 65 + wc + 2] = wv.z;
            Ws[cur][wr * 65 + wc + 3] = wv.w;
        }

        __syncthreads();   // tile (all waves' portions) visible in LDS

        #pragma unroll
        for (int kk = 0; kk < 16; kk += 4) {
            v2f a, b0, b1;
            // A frag 16x4: lanes 0-15 K=(kk,kk+1), lanes 16-31 K=(kk+2,kk+3)
            a.x = As[cur][(mW + l) * 17 + kk + hi * 2 + 0];
            a.y = As[cur][(mW + l) * 17 + kk + hi * 2 + 1];
            // B frag 4x16: same K split, N across lanes
            b0.x = Ws[cur][(kk + hi * 2 + 0) * 65 + nW + l];
            b0.y = Ws[cur][(kk + hi * 2 + 1) * 65 + nW + l];
            b1.x = Ws[cur][(kk + hi * 2 + 0) * 65 + nW + 16 + l];
            b1.y = Ws[cur][(kk + hi * 2 + 1) * 65 + nW + 16 + l];
            c0 = __builtin_amdgcn_wmma_f32_16x16x4_f32(false, a, false, b0,
                                                       (short)0, c0, false, false);
            c1 = __builtin_amdgcn_wmma_f32_16x16x4_f32(false, a, false, b1,
                                                       (short)0, c1, false, false);
        }

        __syncthreads();   // protect buffer reuse before next staging
    }

    // ---- epilogue: bias (+ReLU), masked store ----
    const int col0 = nBlock + nW + l;
    const int col1 = col0 + 16;
    const float bs0 = bias ? bias[col0] : 0.f;
    const float bs1 = bias ? bias[col1] : 0.f;
    #pragma unroll
    for (int r = 0; r < 8; ++r) {
        const int row = mBlock + mW + hi * 8 + r;
        if (row < M) {
            float v0 = c0[r] + bs0;
            float v1 = c1[r] + bs1;
            if (doRelu) { v0 = fmaxf(v0, 0.f); v1 = fmaxf(v1, 0.f); }
            C[(size_t)row * N + col0] = v0;
            C[(size_t)row * N + col1] = v1;
        }
    }
}

// ---------------------------------------------------------------------------
// Fused message kernel: m = relu(x[src] + edge_attr @ We + be);
// aggr[dst] += m  (aggr was pre-seeded with x so it ends up as h = x + sum m).
// One wave per edge, We/be staged in LDS, f32 atomics resolve in L2.
// ---------------------------------------------------------------------------
template <int D>
__global__ __launch_bounds__(256)
void msg_scatter_kernel(const float* __restrict__ x,
                        const int* __restrict__ src, const int* __restrict__ dst,
                        const float* __restrict__ ea,
                        const float* __restrict__ We, const float* __restrict__ be,
                        float* __restrict__ aggr, int E)
{
    constexpr int CPL = D / 32;         // columns per lane (2 or 8)
    __shared__ float sWe[FEDGE * D];
    __shared__ float sBe[D];
    for (int i = threadIdx.x; i < FEDGE * D; i += 256) sWe[i] = We[i];
    for (int i = threadIdx.x; i < D; i += 256)         sBe[i] = be[i];
    __syncthreads();

    const int wid  = threadIdx.x >> 5;
    const int lane = threadIdx.x & 31;
    const int e = blockIdx.x * 8 + wid;
    if (e >= E) return;

    const int s = src[e];
    const int t = dst[e];

    float eav[FEDGE];
    #pragma unroll
    for (int k = 0; k < FEDGE; ++k) eav[k] = ea[(size_t)e * FEDGE + k];

    const float* xrow = x + (size_t)s * D + lane * CPL;
    float acc[CPL];
    #pragma unroll
    for (int j = 0; j < CPL; ++j) acc[j] = xrow[j] + sBe[lane * CPL + j];
    #pragma unroll
    for (int k = 0; k < FEDGE; ++k) {
        const float a = eav[k];
        #pragma unroll
        for (int j = 0; j < CPL; ++j)
            acc[j] = fmaf(a, sWe[k * D + lane * CPL + j], acc[j]);
    }
    float* arow = aggr + (size_t)t * D + lane * CPL;
    #pragma unroll
    for (int j = 0; j < CPL; ++j)
        atomicAdd(&arow[j], fmaxf(acc[j], 0.f));
}

// ---------------------------------------------------------------------------
// Column sums / sum-of-squares over pre-activations (for BatchNorm stats).
// ---------------------------------------------------------------------------
__global__ __launch_bounds__(256)
void col_stats_kernel(const float* __restrict__ pre, int M, float* __restrict__ stats)
{
    const int c  = threadIdx.x;                 // column 0..255
    const int r0 = blockIdx.x * 128;
    const int r1 = (r0 + 128 < M) ? r0 + 128 : M;
    float s = 0.f, sq = 0.f;
    for (int r = r0; r < r1; ++r) {
        const float v = pre[(size_t)r * HDIM + c];
        s += v;
        sq += v * v;
    }
    atomicAdd(&stats[c], s);
    atomicAdd(&stats[HDIM + c], sq);
}

__global__ void bn_finalize_kernel(const float* __restrict__ stats,
                                   const float* __restrict__ gamma,
                                   const float* __restrict__ beta,
                                   float* __restrict__ scaleShift, float invM)
{
    const int c = threadIdx.x;
    const float mu  = stats[c] * invM;
    const float var = stats[HDIM + c] * invM - mu * mu;
    const float sc  = gamma[c] * rsqrtf(var + BN_EPS);
    scaleShift[c]        = sc;
    scaleShift[HDIM + c] = beta[c] - mu * sc;
}

// global_add_pool into z[:, colOff:colOff+256]
__global__ __launch_bounds__(256)
void pool_kernel(const float* __restrict__ h, const int* __restrict__ batch,
                 float* __restrict__ z, int colOff)
{
    const size_t i = (size_t)blockIdx.x * 256 + threadIdx.x;
    if (i >= (size_t)NNODES * HDIM) return;
    const int n = (int)(i >> 8);
    const int c = (int)(i & 255);
    atomicAdd(&z[(size_t)batch[n] * (4 * HDIM) + colOff + c], h[i]);
}

__global__ void zero_kernel(float* __restrict__ p, size_t n)
{
    const size_t i = (size_t)blockIdx.x * 256 + threadIdx.x;
    if (i < n) p[i] = 0.f;
}

// final head matvec: out[g] = hh[g,:] . W2[:,0] + b2
__global__ __launch_bounds__(256)
void head_out_kernel(const float* __restrict__ hh, const float* __restrict__ W2,
                     const float* __restrict__ b2, float* __restrict__ out)
{
    __shared__ float red[256];
    const int g = blockIdx.x;
    const int t = threadIdx.x;
    float s = 0.f;
    #pragma unroll
    for (int k = 0; k < 4; ++k)
        s += hh[(size_t)g * 1024 + t + k * 256] * W2[t + k * 256];
    red[t] = s;
    __syncthreads();
    for (int st = 128; st > 0; st >>= 1) {
        if (t < st) red[t] += red[t + st];
        __syncthreads();
    }
    if (t == 0) out[g] = red[0] + b2[0];
}

// ---------------------------------------------------------------------------
extern "C" void kernel_launch(void* const* d_in, const int* in_sizes, int n_in,
                              void* d_out, int out_size, void* d_ws, size_t ws_size,
                              hipStream_t stream)
{
    (void)in_sizes; (void)n_in; (void)out_size; (void)ws_size;

    const float* x0    = (const float*)d_in[0];
    const int*   eidx  = (const int*)d_in[1];
    const float* eattr = (const float*)d_in[2];
    const int*   batch = (const int*)d_in[3];
    const float* cp[4][8];
    for (int lyr = 0; lyr < 4; ++lyr)
        for (int j = 0; j < 8; ++j)
            cp[lyr][j] = (const float*)d_in[4 + lyr * 8 + j];
    const float* Wh1 = (const float*)d_in[36];
    const float* bh1 = (const float*)d_in[37];
    const float* Wh2 = (const float*)d_in[38];
    const float* bh2 = (const float*)d_in[39];
    float* out = (float*)d_out;

    // workspace layout
    float* ws   = (float*)d_ws;
    const size_t nb = (size_t)NNODES * HDIM;
    float* bufA  = ws;                         // h = x + aggr      [N,256]
    float* bufB  = bufA + nb;                  // pre-activations   [N,256]
    float* bufC  = bufB + nb;                  // layer output      [N,256]
    float* z     = bufC + nb;                  // pooled            [256,1024]
    float* hh    = z + (size_t)NGRAPH * 1024;  // head hidden       [256,1024]
    float* stats = hh + (size_t)NGRAPH * 1024; // [512]
    float* ss    = stats + 512;                // scale/shift [512]

    const int* srcI = eidx;
    const int* dstI = eidx + NEDGES;

    zero_kernel<<<(NGRAPH * 1024 + 255) / 256, 256, 0, stream>>>(z, (size_t)NGRAPH * 1024);

    const float* xcur = x0;
    for (int lyr = 0; lyr < 4; ++lyr) {
        const int din = (lyr == 0) ? FNODE : HDIM;
        const float* We = cp[lyr][0]; const float* be = cp[lyr][1];
        const float* W1 = cp[lyr][2]; const float* b1 = cp[lyr][3];
        const float* ga = cp[lyr][4]; const float* bt = cp[lyr][5];
        const float* W2 = cp[lyr][6]; const float* b2 = cp[lyr][7];

        // seed aggregate buffer with x (h = x + sum of messages)
        hipMemcpyAsync(bufA, xcur, (size_t)NNODES * din * sizeof(float),
                       hipMemcpyDeviceToDevice, stream);

        const int mblocks = (NEDGES + 7) / 8;
        if (din == FNODE)
            msg_scatter_kernel<FNODE><<<mblocks, 256, 0, stream>>>(
                xcur, srcI, dstI, eattr, We, be, bufA, NEDGES);
        else
            msg_scatter_kernel<HDIM><<<mblocks, 256, 0, stream>>>(
                xcur, srcI, dstI, eattr, We, be, bufA, NEDGES);

        // GEMM1 (async-staged): pre = h @ W1 + b1
        dim3 g1(HDIM / 64, (NNODES + 63) / 64);
        gemm_wmma_kernel<false><<<g1, 256, 0, stream>>>(
            bufA, W1, b1, nullptr, nullptr, bufB, NNODES, din, HDIM, 0);

        // BatchNorm statistics + fused-affine coefficients
        zero_kernel<<<2, 256, 0, stream>>>(stats, 512);
        col_stats_kernel<<<(NNODES + 127) / 128, 256, 0, stream>>>(bufB, NNODES, stats);
        bn_finalize_kernel<<<1, 256, 0, stream>>>(stats, ga, bt, ss, 1.0f / NNODES);

        // GEMM2 (BN+ReLU fused into A staging): hout = relu(BNrelu(pre) @ W2 + b2)
        gemm_wmma_kernel<true><<<g1, 256, 0, stream>>>(
            bufB, W2, b2, ss, ss + HDIM, bufC, NNODES, HDIM, HDIM, 1);

        // global_add_pool into z columns [lyr*256, lyr*256+256)
        pool_kernel<<<(int)(((size_t)NNODES * HDIM + 255) / 256), 256, 0, stream>>>(
            bufC, batch, z, lyr * HDIM);

        xcur = bufC;
    }

    // head: hh = relu(z @ Wh1 + bh1); out = hh @ Wh2 + bh2
    dim3 gh(1024 / 64, (NGRAPH + 63) / 64);
    gemm_wmma_kernel<false><<<gh, 256, 0, stream>>>(
        z, Wh1, bh1, nullptr, nullptr, hh, NGRAPH, 1024, 1024, 1);
    head_out_kernel<<<NGRAPH, 256, 0, stream>>>(hh, Wh2, bh2, out);
}